// PrismBrain_66949950210719
// MI455X (gfx1250) — compile-verified
//
#include <hip/hip_runtime.h>
#include <math.h>

typedef float v2f __attribute__((ext_vector_type(2)));
typedef float v4f __attribute__((ext_vector_type(4)));
typedef float v8f __attribute__((ext_vector_type(8)));

#define N_CELLS 4096
#define CHUNK   8
#define HID     1024
#define WAVES_PER_BLOCK 8

// Branch-free tanh on the native transcendental pipe:
// tanh(v) = 1 - 2/(exp(2v)+1)  ->  v_mul, v_exp_f32, v_add, v_rcp_f32, v_fma.
// No EXEC-mask manipulation (keeps EXEC all-ones around the WMMA stream),
// saturates to +-1 for large |v| (exp -> inf/0).
__device__ __forceinline__ float fast_tanh(float v) {
    float e = __expf(2.0f * v);
    return 1.0f - 2.0f * __builtin_amdgcn_rcpf(e + 1.0f);
}

// One cell per wave32. GEMV1 goes through v_wmma_f32_16x16x4_f32 with x
// replicated across the 16 A-rows (rows of D are then identical, so every
// lane's C[0] is the pre-activation of hidden column 16t + (lane&15)).
// Two 16-wide hidden chunks per iteration -> lane l owns hidden row 32r+l,
// giving perfectly coalesced, duplicate-free streaming of W2.
__global__ __launch_bounds__(256) void cellmlp_wmma_kernel(
    const float* __restrict__ x_all,
    const float* __restrict__ W1,
    const float* __restrict__ b1,
    const float* __restrict__ W2,
    const float* __restrict__ b2,
    float* __restrict__ out)
{
    const int lane = threadIdx.x & 31;
    const int wave = threadIdx.x >> 5;
    const int cell = blockIdx.x * WAVES_PER_BLOCK + wave;

    const float* xg  = x_all + cell * CHUNK;
    const float* W1n = W1 + (size_t)cell * CHUNK * HID;   // [8][1024] row-major
    const float* b1n = b1 + (size_t)cell * HID;
    const float* W2n = W2 + (size_t)cell * HID * CHUNK;   // [1024][8] row-major

    // Per-cell input chunk (uniform across the wave).
    float x[CHUNK];
#pragma unroll
    for (int i = 0; i < CHUNK; ++i) x[i] = xg[i];

    const int col  = lane & 15;   // N index of this lane in B/C/D
    const int hi   = lane >> 4;   // which half-wave
    const int crow = hi * 2;      // K rows this half supplies (A/B layout)

    // A operands: x replicated into all 16 M-rows.
    // A 16x4 f32 layout: lanes 0-15 VGPR0=K0,VGPR1=K1; lanes 16-31 K2,K3.
    v2f A0, A1;
    A0[0] = x[crow];     A0[1] = x[crow + 1];      // K-slice c=0..3
    A1[0] = x[4 + crow]; A1[1] = x[4 + crow + 1];  // K-slice c=4..7

    v8f yacc = {0.f, 0.f, 0.f, 0.f, 0.f, 0.f, 0.f, 0.f};

    for (int r = 0; r < HID / 32; ++r) {
        const int j0 = 32 * r + col;  // hidden column, chunk 0
        const int j1 = j0 + 16;       // hidden column, chunk 1

        // B operands: W1 tile, B 4x16 layout mirrors A
        // (lanes 0-15: K0 in v[0], K1 in v[1]; lanes 16-31: K2, K3).
        v2f B00, B01, B10, B11;
        B00[0] = __builtin_nontemporal_load(W1n + (crow    ) * HID + j0);
        B00[1] = __builtin_nontemporal_load(W1n + (crow + 1) * HID + j0);
        B01[0] = __builtin_nontemporal_load(W1n + (4 + crow    ) * HID + j0);
        B01[1] = __builtin_nontemporal_load(W1n + (4 + crow + 1) * HID + j0);
        B10[0] = __builtin_nontemporal_load(W1n + (crow    ) * HID + j1);
        B10[1] = __builtin_nontemporal_load(W1n + (crow + 1) * HID + j1);
        B11[0] = __builtin_nontemporal_load(W1n + (4 + crow    ) * HID + j1);
        B11[1] = __builtin_nontemporal_load(W1n + (4 + crow + 1) * HID + j1);

        v8f c0 = {0.f, 0.f, 0.f, 0.f, 0.f, 0.f, 0.f, 0.f};
        v8f c1 = {0.f, 0.f, 0.f, 0.f, 0.f, 0.f, 0.f, 0.f};
        // D = A*B + C, accumulate both K=4 slices of the 8-deep reduction.
        c0 = __builtin_amdgcn_wmma_f32_16x16x4_f32(false, A0, false, B00,
                                                   (short)0, c0, false, false);
        c0 = __builtin_amdgcn_wmma_f32_16x16x4_f32(false, A1, false, B01,
                                                   (short)0, c0, false, false);
        c1 = __builtin_amdgcn_wmma_f32_16x16x4_f32(false, A0, false, B10,
                                                   (short)0, c1, false, false);
        c1 = __builtin_amdgcn_wmma_f32_16x16x4_f32(false, A1, false, B11,
                                                   (short)0, c1, false, false);

        // Every lane's c[0] holds the pre-activation of its N column
        // (all D rows identical). Half-waves take different chunks so that
        // lane l owns hidden row = 32r + l.
        const int row = 32 * r + lane;
        const float pre = hi ? c1[0] : c0[0];
        const float h = fast_tanh(pre + __builtin_nontemporal_load(b1n + row));

        // GEMV2 partial: fully coalesced 1KB/wave stream of W2 rows.
        const v4f w2a = __builtin_nontemporal_load(
            (const v4f*)(W2n + (size_t)row * CHUNK));
        const v4f w2b = __builtin_nontemporal_load(
            (const v4f*)(W2n + (size_t)row * CHUNK + 4));
        yacc[0] += h * w2a[0];
        yacc[1] += h * w2a[1];
        yacc[2] += h * w2a[2];
        yacc[3] += h * w2a[3];
        yacc[4] += h * w2b[0];
        yacc[5] += h * w2b[1];
        yacc[6] += h * w2b[2];
        yacc[7] += h * w2b[3];

        if (r + 1 < HID / 32) {
            // Speculative streaming prefetch of the next W2 block.
            __builtin_prefetch(W2n + (size_t)(row + 32) * CHUNK, 0, 1);
        }
    }

    // Butterfly reduction across the wave: every lane ends with full y[0..7].
#pragma unroll
    for (int off = 16; off >= 1; off >>= 1) {
#pragma unroll
        for (int c = 0; c < CHUNK; ++c)
            yacc[c] += __shfl_xor(yacc[c], off, 32);
    }

    if (lane == 0) {
#pragma unroll
        for (int c = 0; c < CHUNK; ++c)
            out[(size_t)cell * CHUNK + c] = yacc[c] + b2[(size_t)cell * CHUNK + c];
    }
}

extern "C" void kernel_launch(void* const* d_in, const int* in_sizes, int n_in,
                              void* d_out, int out_size, void* d_ws, size_t ws_size,
                              hipStream_t stream) {
    const float* x  = (const float*)d_in[0];
    const float* W1 = (const float*)d_in[1];
    const float* b1 = (const float*)d_in[2];
    const float* W2 = (const float*)d_in[3];
    const float* b2 = (const float*)d_in[4];
    float* out = (float*)d_out;

    dim3 grid(N_CELLS / WAVES_PER_BLOCK);  // 512 blocks
    dim3 block(WAVES_PER_BLOCK * 32);      // 8 waves of 32
    cellmlp_wmma_kernel<<<grid, block, 0, stream>>>(x, W1, b1, W2, b2, out);
}